// Cluster_47614007444067
// MI455X (gfx1250) — compile-verified
//
#include <hip/hip_runtime.h>
#include <math.h>

// CDNA5 / gfx1250 implementation of the gathered diagonal-GMM log-likelihood.
//
// Strategy (see analysis):
//   Sigma-weighted squared distance expanded into two batched dot products
//   (q1 = <xs^2, sigma^2>, q2 = <xs, mu*sigma^2>) computed with
//   V_WMMA_F32_16X16X4_F32 (16 batch rows per tile, per-batch class vectors
//   as B columns, diagonal harvest), plus per-(class,center) constants
//   t3 = <mu^2, sigma^2>, log_det = sum log(sigma), and logsumexp(class_pi).
//   f32 WMMA keeps numerics aligned with the f32 reference (values ~1e3).

typedef float v2f __attribute__((ext_vector_type(2)));
typedef float v8f __attribute__((ext_vector_type(8)));

#define NCLASS   100
#define NCENTER  4
#define HID      512
#define BATCH    512
#define INV_GAMA 0.1f

// ---------------------------------------------------------------------------
// Kernel 1: per-(class,center) constants  t3[ck] = sum mu^2*sigma^2,
//                                         ld[ck] = sum log(sigma)
// ---------------------------------------------------------------------------
__global__ __launch_bounds__(64) void precompute_ck(
    const float* __restrict__ class_mu, const float* __restrict__ class_sigma,
    float* __restrict__ t3, float* __restrict__ ld) {
  __shared__ float s1[64];
  __shared__ float s2[64];
  const int ck = blockIdx.x;          // 0..399
  const int t  = threadIdx.x;         // 0..63
  const float* mu = class_mu    + (size_t)ck * HID;
  const float* sg = class_sigma + (size_t)ck * HID;
  float a3 = 0.0f, al = 0.0f;
  for (int h = t; h < HID; h += 64) {
    const float m = mu[h];
    const float s = sg[h];
    a3 += m * m * s * s;
    al += logf(s);
  }
  s1[t] = a3; s2[t] = al;
  __syncthreads();
  for (int off = 32; off > 0; off >>= 1) {
    if (t < off) { s1[t] += s1[t + off]; s2[t] += s2[t + off]; }
    __syncthreads();
  }
  if (t == 0) { t3[ck] = s1[0]; ld[ck] = s2[0]; }
}

// ---------------------------------------------------------------------------
// Kernel 2: lse[c] = logsumexp(class_pi[c, 0..3])
// ---------------------------------------------------------------------------
__global__ __launch_bounds__(128) void precompute_lse(
    const float* __restrict__ class_pi, float* __restrict__ lse) {
  const int c = blockIdx.x * blockDim.x + threadIdx.x;
  if (c < NCLASS) {
    const float p0 = class_pi[c * NCENTER + 0];
    const float p1 = class_pi[c * NCENTER + 1];
    const float p2 = class_pi[c * NCENTER + 2];
    const float p3 = class_pi[c * NCENTER + 3];
    const float m  = fmaxf(fmaxf(p0, p1), fmaxf(p2, p3));
    lse[c] = m + logf(expf(p0 - m) + expf(p1 - m) + expf(p2 - m) + expf(p3 - m));
  }
}

// ---------------------------------------------------------------------------
// Kernel 3: WMMA core. One block = 16 batch rows, 4 waves split H=512 into
// 128-wide slices. Each wave runs 8 f32 WMMA accumulators (4 centers x
// {q1,q2}), harvests accumulator diagonals, cross-wave reduce in LDS, then
// 16 threads finish the per-batch max-over-centers log-likelihood.
// ---------------------------------------------------------------------------
__global__ __launch_bounds__(128) void cluster_ll_wmma(
    const float* __restrict__ x, const int* __restrict__ y,
    const float* __restrict__ mu_base,
    const float* __restrict__ class_mu, const float* __restrict__ class_sigma,
    const float* __restrict__ class_pi,
    const float* __restrict__ t3, const float* __restrict__ ld,
    const float* __restrict__ lse, float* __restrict__ ll) {
  __shared__ float s_p1[4][NCENTER][16];   // per-wave diag partials, term q1
  __shared__ float s_p2[4][NCENTER][16];   // per-wave diag partials, term q2
  __shared__ float s_q1[NCENTER][16];
  __shared__ float s_q2[NCENTER][16];

  const int tid  = threadIdx.x;
  const int wv   = tid >> 5;       // wave 0..3
  const int lane = tid & 31;
  const int half = lane >> 4;      // K-pair selector (ISA f32 A/B layout)
  const int l    = lane & 15;      // A row index == B column index == batch-in-tile
  const int row_base = blockIdx.x * 16;
  const int b_idx = row_base + l;
  const int cls   = y[b_idx];

  const float* xrow = x + (size_t)b_idx * HID;
  const float* muc  = class_mu    + (size_t)cls * NCENTER * HID;
  const float* sgc  = class_sigma + (size_t)cls * NCENTER * HID;

  v8f c1[NCENTER] = {};   // q1 accumulators:  xs^2 . sigma^2
  v8f c2[NCENTER] = {};   // q2 accumulators:  xs   . mu*sigma^2

  const int h_lo = wv * (HID / 4);          // 128 h-values per wave
  for (int h0 = h_lo; h0 < h_lo + HID / 4; h0 += 4) {
    const int hoff = h0 + 2 * half;         // this lane's K-pair offset
    const float2 xv = *(const float2*)(xrow + hoff);
    const float2 mb = *(const float2*)(mu_base + hoff);
    const float xs0 = (xv.x - mb.x) * INV_GAMA;
    const float xs1 = (xv.y - mb.y) * INV_GAMA;
    v2f a1; a1[0] = xs0 * xs0; a1[1] = xs1 * xs1;   // A tile for q1
    v2f a2; a2[0] = xs0;       a2[1] = xs1;         // A tile for q2
#pragma unroll
    for (int k = 0; k < NCENTER; ++k) {
      const float2 sg = *(const float2*)(sgc + k * HID + hoff);
      const float2 mu = *(const float2*)(muc + k * HID + hoff);
      v2f b1; b1[0] = sg.x * sg.x;  b1[1] = sg.y * sg.y;   // sigma^2
      v2f b2; b2[0] = mu.x * b1[0]; b2[1] = mu.y * b1[1];  // mu*sigma^2
      c1[k] = __builtin_amdgcn_wmma_f32_16x16x4_f32(
          false, a1, false, b1, (short)0, c1[k], false, false);
      c2[k] = __builtin_amdgcn_wmma_f32_16x16x4_f32(
          false, a2, false, b2, (short)0, c2[k], false, false);
    }
  }

  // Diagonal harvest. D[j,j]: j<8 -> lane j, VGPR j; j>=8 -> lane 16+j, VGPR j-8.
  const bool hasdiag = ((l >> 3) == half);
  const int  rsel    = l & 7;
  if (hasdiag) {
#pragma unroll
    for (int k = 0; k < NCENTER; ++k) {
      float d1 = 0.0f, d2 = 0.0f;
#pragma unroll
      for (int r = 0; r < 8; ++r) {
        d1 = (r == rsel) ? c1[k][r] : d1;
        d2 = (r == rsel) ? c2[k][r] : d2;
      }
      s_p1[wv][k][l] = d1;
      s_p2[wv][k][l] = d2;
    }
  }
  __syncthreads();

  if (tid < 64) {  // reduce the 4 wave partials for each (center k, row j)
    const int j = tid & 15, k = tid >> 4;
    s_q1[k][j] = s_p1[0][k][j] + s_p1[1][k][j] + s_p1[2][k][j] + s_p1[3][k][j];
    s_q2[k][j] = s_p2[0][k][j] + s_p2[1][k][j] + s_p2[2][k][j] + s_p2[3][k][j];
  }
  __syncthreads();

  if (tid < 16) {
    const int b = row_base + tid;
    const int c = y[b];
    const float KONST = -0.5f * (float)HID * 1.8378770664093453f; // -H/2*log(2pi)
    float best = -3.402823466e38f;
#pragma unroll
    for (int k = 0; k < NCENTER; ++k) {
      const int ck = c * NCENTER + k;
      const float lexp  = -0.5f * (s_q1[k][tid] - 2.0f * s_q2[k][tid] + t3[ck]);
      const float lp    = lexp + ld[ck] + KONST;
      const float score = (class_pi[ck] - lse[c]) + lp;
      best = fmaxf(best, score);
    }
    ll[b] = best;
  }
}

// ---------------------------------------------------------------------------
// Kernel 4: deterministic mean over the 512 per-batch log-likelihoods.
// ---------------------------------------------------------------------------
__global__ __launch_bounds__(256) void reduce_mean(
    const float* __restrict__ ll, float* __restrict__ out) {
  __shared__ float s[256];
  const int t = threadIdx.x;
  s[t] = ll[t] + ll[t + 256];
  __syncthreads();
  for (int off = 128; off > 0; off >>= 1) {
    if (t < off) s[t] += s[t + off];
    __syncthreads();
  }
  if (t == 0) out[0] = s[0] * (1.0f / (float)BATCH);
}

// ---------------------------------------------------------------------------
extern "C" void kernel_launch(void* const* d_in, const int* in_sizes, int n_in,
                              void* d_out, int out_size, void* d_ws, size_t ws_size,
                              hipStream_t stream) {
  (void)in_sizes; (void)n_in; (void)out_size; (void)ws_size;
  const float* x           = (const float*)d_in[0];
  const int*   y           = (const int*)  d_in[1];   // integer input -> const int*
  const float* mu_base     = (const float*)d_in[2];
  const float* class_mu    = (const float*)d_in[3];
  const float* class_sigma = (const float*)d_in[4];
  const float* class_pi    = (const float*)d_in[5];

  float* ws  = (float*)d_ws;
  float* t3  = ws;          // 400 floats: sum mu^2*sigma^2 per (c,k)
  float* ld  = ws + 400;    // 400 floats: sum log(sigma)   per (c,k)
  float* lse = ws + 800;    // 100 floats: logsumexp(class_pi[c,:])
  float* ll  = ws + 912;    // 512 floats: per-batch log-likelihood

  precompute_ck<<<NCLASS * NCENTER, 64, 0, stream>>>(class_mu, class_sigma, t3, ld);
  precompute_lse<<<1, 128, 0, stream>>>(class_pi, lse);
  cluster_ll_wmma<<<BATCH / 16, 128, 0, stream>>>(
      x, y, mu_base, class_mu, class_sigma, class_pi, t3, ld, lse, ll);
  reduce_mean<<<1, 256, 0, stream>>>(ll, (float*)d_out);
}